// TaskSpecificGIMANTest_33492154974677
// MI455X (gfx1250) — compile-verified
//
#include <hip/hip_runtime.h>
#include <cstddef>

#define DEV __device__ __forceinline__

typedef float v2f __attribute__((ext_vector_type(2)));
typedef float v8f __attribute__((ext_vector_type(8)));

static constexpr int Bv = 8, Nv = 1024, SPv = 256, GEv = 8, TEv = 64;
static constexpr int Ev = 32, Hv = 2, HIDv = 32, HDv = 16;
static constexpr int PT = 70;   // LDS row pitch (floats): even (8B-aligned b64
                                // fragment loads) and == 6 mod 64 banks
                                // (conflict-free half-wave fragment fetch).

DEV v8f wmma4(v2f a, v2f b, v8f c) {
  // D = A(16x4 f32) * B(4x16 f32) + C(16x16 f32)
  return __builtin_amdgcn_wmma_f32_16x16x4_f32(false, a, false, b, (short)0, c, false, false);
}

DEV float wsum(float v) {
#pragma unroll
  for (int m = 16; m >= 1; m >>= 1) v += __shfl_xor(v, m, 32);
  return v;
}

// LayerNorm across the 32 lanes of a wave (one feature per lane).
DEV float wave_ln(float v, float g, float be) {
  float mu  = wsum(v) * (1.0f / 32.0f);
  float t   = v - mu;
  float var = wsum(t * t) * (1.0f / 32.0f);
  return t * rsqrtf(var + 1e-5f) * g + be;
}

// ---------------------------------------------------------------------------
// K1: embeds (spatial/genomic/temporal) + LN + ReLU, GAT input proj h,
//     rank-1 attention logits s,d.  One wave per (b,n) row.
// ---------------------------------------------------------------------------
__global__ void __launch_bounds__(256) k_embed(
    const float* __restrict__ spatial, const float* __restrict__ genomic,
    const float* __restrict__ temporal,
    const float* __restrict__ spW, const float* __restrict__ spb,
    const float* __restrict__ spg, const float* __restrict__ spbe,
    const float* __restrict__ geW, const float* __restrict__ geb,
    const float* __restrict__ geg, const float* __restrict__ gebe,
    const float* __restrict__ teW, const float* __restrict__ teb,
    const float* __restrict__ teg, const float* __restrict__ tebe,
    const float* __restrict__ gatW, const float* __restrict__ gata,
    float* __restrict__ h_ws, float* __restrict__ s_ws, float* __restrict__ d_ws) {
  __shared__ float sx[8][96];
  const int w = threadIdx.x >> 5, lane = threadIdx.x & 31;
  const int row = blockIdx.x * 8 + w;
  const int b = row >> 10;

  // spatial embed (K=256)
  float acc = spb[lane];
  {
    const float* xr = spatial + (size_t)row * SPv;
    const float* wr = spW + lane * SPv;
    for (int k = 0; k < SPv; ++k) acc += xr[k] * wr[k];
  }
  float spE = fmaxf(wave_ln(acc, spg[lane], spbe[lane]), 0.0f);

  // genomic embed (K=8), broadcast over n
  acc = geb[lane];
  {
    const float* xr = genomic + b * GEv;
    const float* wr = geW + lane * GEv;
    for (int k = 0; k < GEv; ++k) acc += xr[k] * wr[k];
  }
  float geE = fmaxf(wave_ln(acc, geg[lane], gebe[lane]), 0.0f);

  // temporal embed (K=64)
  acc = teb[lane];
  {
    const float* xr = temporal + (size_t)row * TEv;
    const float* wr = teW + lane * TEv;
    for (int k = 0; k < TEv; ++k) acc += xr[k] * wr[k];
  }
  float teE = fmaxf(wave_ln(acc, teg[lane], tebe[lane]), 0.0f);

  sx[w][lane] = spE; sx[w][32 + lane] = geE; sx[w][64 + lane] = teE;
  __syncthreads();

  const float a_src = gata[lane];
  const float a_dst = gata[32 + lane];
#pragma unroll
  for (int hh = 0; hh < Hv; ++hh) {
    const int o = hh * HIDv + lane;
    float hv = 0.0f;
    const float* wo = gatW + o * 96;
    for (int k = 0; k < 96; ++k) hv += sx[w][k] * wo[k];
    h_ws[(size_t)row * 64 + o] = hv;
    float sv = wsum(hv * a_src);
    float dv = wsum(hv * a_dst);
    if (lane == 0) { s_ws[row * Hv + hh] = sv; d_ws[row * Hv + hh] = dv; }
  }
}

// ---------------------------------------------------------------------------
// K2: GAT aggregation.  alpha_ij = adj_ij * exp(lrelu(s_i + d_j)), then
//     h_out = (alpha @ h) / rowsum  via  V_WMMA_F32_16X16X4_F32.
//     Block = 4 waves, each owns a 16-row i-tile; shared (b,h) j-chunks.
//     All WMMA fragments are single ds_load_b64 (contiguous, 8B-aligned).
// ---------------------------------------------------------------------------
__global__ void __launch_bounds__(128) k_gat(
    const float* __restrict__ h_ws, const float* __restrict__ s_ws,
    const float* __restrict__ d_ws, const float* __restrict__ adj,
    float* __restrict__ hcat) {
  const int blk = blockIdx.x;            // B*H*(N/64) = 256
  const int b = blk >> 5;
  const int rem = blk & 31;
  const int hh = rem >> 4;
  const int it = rem & 15;
  const int w = threadIdx.x >> 5, lane = threadIdx.x & 31;
  const int i0 = it * 64 + w * 16;
  const int l16 = lane & 15;
  const bool hi = lane >= 16;

  __shared__ __align__(16) float sBT[32][PT];      // h_j chunk transposed [f][j]
  __shared__ float sD[64];                         // d_j chunk
  __shared__ __align__(16) float sAl[4][16][PT];   // per-wave alpha [row][j]
  __shared__ float sDen[4][16];

  const int rloc = lane >> 1;
  const int jbase = (lane & 1) * 32;
  const float s_r = s_ws[(b * Nv + i0 + rloc) * Hv + hh];

  v8f c0 = {}; v8f c1 = {};
  float rs = 0.0f;

  for (int jc = 0; jc < Nv; jc += 64) {
    __syncthreads();
    // coalesced in f (lane-contiguous); LDS pitch keeps stores conflict-free
    for (int idx = threadIdx.x; idx < 64 * 32; idx += 128) {
      const int jl = idx >> 5, f = idx & 31;
      sBT[f][jl] = h_ws[(size_t)(b * Nv + jc + jl) * 64 + hh * 32 + f];
    }
    if (threadIdx.x < 64)
      sD[threadIdx.x] = d_ws[(b * Nv + jc + threadIdx.x) * Hv + hh];
    __syncthreads();

    // masked exp(leaky_relu) tile; accumulate per-row partial sums
    const float* adjrow = adj + (size_t)(i0 + rloc) * Nv + jc + jbase;
#pragma unroll 4
    for (int jj = 0; jj < 32; ++jj) {
      const float aj = adjrow[jj];
      const float x = s_r + sD[jbase + jj];
      const float e = (x > 0.0f) ? x : 0.2f * x;
      const float val = (aj != 0.0f) ? __expf(e) : 0.0f;
      sAl[w][rloc][jbase + jj] = val;
      rs += val;
    }
    __syncthreads();

    if (jc + 64 < Nv)
      __builtin_prefetch(h_ws + (size_t)(b * Nv + jc + 64) * 64, 0, 1);

#pragma unroll
    for (int ks = 0; ks < 64; ks += 4) {
      const int k0 = ks + (hi ? 2 : 0);
      const v2f A  = *(const v2f*)&sAl[w][l16][k0];
      const v2f B0 = *(const v2f*)&sBT[l16][k0];
      const v2f B1 = *(const v2f*)&sBT[16 + l16][k0];
      c0 = wmma4(A, B0, c0);
      c1 = wmma4(A, B1, c1);
    }
  }

  rs += __shfl_xor(rs, 1, 32);
  if ((lane & 1) == 0) sDen[w][rloc] = rs;
  __syncthreads();

#pragma unroll
  for (int r = 0; r < 8; ++r) {
    const int m = r + (hi ? 8 : 0);
    const float inv = 1.0f / sDen[w][m];
    const size_t base = (size_t)(b * Nv + i0 + m) * 64 + hh * 32;
    hcat[base + l16]      = c0[r] * inv;
    hcat[base + 16 + l16] = c1[r] * inv;
  }
}

// ---------------------------------------------------------------------------
// K3: g = LN(hcat @ oW^T + ob), then fused qkv projection, stored head-major.
// ---------------------------------------------------------------------------
__global__ void __launch_bounds__(256) k_gproj(
    const float* __restrict__ hcat, const float* __restrict__ oW,
    const float* __restrict__ ob, const float* __restrict__ gg,
    const float* __restrict__ gbe, const float* __restrict__ iW,
    const float* __restrict__ ib,
    float* __restrict__ q_ws, float* __restrict__ k_ws, float* __restrict__ v_ws) {
  __shared__ float sg[8][32];
  const int w = threadIdx.x >> 5, lane = threadIdx.x & 31;
  const int row = blockIdx.x * 8 + w;
  const int b = row >> 10, n = row & 1023;

  float acc = ob[lane];
  {
    const float* hr = hcat + (size_t)row * 64;
    const float* wr = oW + lane * 64;
    for (int k = 0; k < 64; ++k) acc += hr[k] * wr[k];
  }
  sg[w][lane] = wave_ln(acc, gg[lane], gbe[lane]);
  __syncthreads();

  const int head = lane >> 4, hd = lane & 15;
#pragma unroll
  for (int t = 0; t < 3; ++t) {
    const int o = t * 32 + lane;
    float a2 = ib[o];
    const float* w2 = iW + o * 32;
    for (int k = 0; k < 32; ++k) a2 += sg[w][k] * w2[k];
    float* dst = (t == 0) ? q_ws : (t == 1 ? k_ws : v_ws);
    dst[((size_t)(b * Hv + head) * Nv + n) * HDv + hd] = a2;
  }
}

// ---------------------------------------------------------------------------
// K4: MHA per (b, head, 16-row i-tile):  S = qk^T/4 (WMMA, K=16), P = exp(S),
//     O = (P @ V)/rowsum (WMMA, K=64).  4 waves/block share K/V chunks.
//     V is stored transposed in LDS so PV B-fragments are contiguous b64 loads.
// ---------------------------------------------------------------------------
__global__ void __launch_bounds__(128) k_attn(
    const float* __restrict__ q_ws, const float* __restrict__ k_ws,
    const float* __restrict__ v_ws, float* __restrict__ o_ws) {
  const int blk = blockIdx.x;            // B*H*(N/64) = 256
  const int b = blk >> 5;
  const int rem = blk & 31;
  const int hh = rem >> 4;
  const int it = rem & 15;
  const int w = threadIdx.x >> 5, lane = threadIdx.x & 31;
  const int i0b = it * 64;
  const int bh = b * Hv + hh;
  const int l16 = lane & 31 & 15;
  const bool hi = lane >= 16;

  __shared__ __align__(16) float sQ[64][PT];       // [i][d]
  __shared__ __align__(16) float sK[64][PT];       // [j][d]
  __shared__ __align__(16) float sVT[16][PT];      // V transposed [d][j]
  __shared__ __align__(16) float sP[4][16][PT];    // per-wave P [row][j]
  __shared__ float sDen[4][16];

  for (int idx = threadIdx.x; idx < 64 * 16; idx += 128) {
    const int r = idx >> 4, dd = idx & 15;
    sQ[r][dd] = q_ws[((size_t)bh * Nv + i0b + r) * HDv + dd];
  }

  v8f O = {};
  float rs = 0.0f;
  const int rloc = lane >> 1;
  const int jbase = (lane & 1) * 32;

  for (int jc = 0; jc < Nv; jc += 64) {
    __syncthreads();
    for (int idx = threadIdx.x; idx < 64 * 16; idx += 128) {
      const int r = idx >> 4, dd = idx & 15;
      sK[r][dd]  = k_ws[((size_t)bh * Nv + jc + r) * HDv + dd];
      sVT[dd][r] = v_ws[((size_t)bh * Nv + jc + r) * HDv + dd];
    }
    __syncthreads();

    // S tiles: 16 x 64, contraction K = HD = 16
    v8f s0 = {}, s1 = {}, s2 = {}, s3 = {};
#pragma unroll
    for (int ks = 0; ks < 16; ks += 4) {
      const int k0 = ks + (hi ? 2 : 0);
      const v2f A = *(const v2f*)&sQ[w * 16 + l16][k0];
      s0 = wmma4(A, *(const v2f*)&sK[l16][k0],      s0);
      s1 = wmma4(A, *(const v2f*)&sK[16 + l16][k0], s1);
      s2 = wmma4(A, *(const v2f*)&sK[32 + l16][k0], s2);
      s3 = wmma4(A, *(const v2f*)&sK[48 + l16][k0], s3);
    }
    // P = exp(S/4), route C-layout -> LDS for A-fragment reuse + row sums
#pragma unroll
    for (int r = 0; r < 8; ++r) {
      const int m = r + (hi ? 8 : 0);
      sP[w][m][l16]      = __expf(s0[r] * 0.25f);
      sP[w][m][16 + l16] = __expf(s1[r] * 0.25f);
      sP[w][m][32 + l16] = __expf(s2[r] * 0.25f);
      sP[w][m][48 + l16] = __expf(s3[r] * 0.25f);
    }
    __syncthreads();
#pragma unroll 4
    for (int jj = 0; jj < 32; ++jj) rs += sP[w][rloc][jbase + jj];

    // O += P @ V   (contraction K = 64)
#pragma unroll
    for (int ks = 0; ks < 64; ks += 4) {
      const int k0 = ks + (hi ? 2 : 0);
      const v2f A = *(const v2f*)&sP[w][l16][k0];
      const v2f B = *(const v2f*)&sVT[l16][k0];
      O = wmma4(A, B, O);
    }
  }

  rs += __shfl_xor(rs, 1, 32);
  if ((lane & 1) == 0) sDen[w][rloc] = rs;
  __syncthreads();

#pragma unroll
  for (int r = 0; r < 8; ++r) {
    const int m = r + (hi ? 8 : 0);
    const float inv = 1.0f / sDen[w][m];
    o_ws[((size_t)(b * Nv + i0b + w * 16 + m)) * 32 + hh * HDv + l16] = O[r] * inv;
  }
}

// ---------------------------------------------------------------------------
// K5: output projection + sum over N -> rep[b][e]
// ---------------------------------------------------------------------------
__global__ void __launch_bounds__(256) k_oproj(
    const float* __restrict__ o_ws, const float* __restrict__ oW,
    const float* __restrict__ ob, float* __restrict__ rep) {
  const int w = threadIdx.x >> 5, lane = threadIdx.x & 31;
  const int row = blockIdx.x * 8 + w;
  const int b = row >> 10;
  float acc = ob[lane];
  const float* orow = o_ws + (size_t)row * 32;
  const float* wr = oW + lane * 32;
  for (int k = 0; k < 32; ++k) acc += orow[k] * wr[k];
  atomicAdd(&rep[b * 32 + lane], acc);
}

// ---------------------------------------------------------------------------
// K6: the two prediction towers (wave per batch element).
// ---------------------------------------------------------------------------
struct TowerP {
  const float *W1, *b1, *g1, *be1, *W2, *b2, *g2, *be2, *W3, *b3;
};

__global__ void __launch_bounds__(256) k_tower(
    const float* __restrict__ rep, TowerP m, TowerP c, float* __restrict__ out) {
  __shared__ float h1s[8][64];
  const int w = threadIdx.x >> 5, lane = threadIdx.x & 31; // w == batch index

  for (int t = 0; t < 2; ++t) {
    const TowerP p = (t == 0) ? m : c;
    // stage 1: 64 dims (lane handles j and j+32), LN over 64, relu
    float a0 = p.b1[lane], a1 = p.b1[lane + 32];
    const float* r0 = rep + w * 32;
    for (int k = 0; k < 32; ++k) {
      const float rv = r0[k];
      a0 += rv * p.W1[lane * 32 + k];
      a1 += rv * p.W1[(lane + 32) * 32 + k];
    }
    const float mu = wsum(a0 + a1) * (1.0f / 64.0f);
    const float t0 = a0 - mu, t1 = a1 - mu;
    const float var = wsum(t0 * t0 + t1 * t1) * (1.0f / 64.0f);
    const float inv = rsqrtf(var + 1e-5f);
    h1s[w][lane]      = fmaxf(t0 * inv * p.g1[lane] + p.be1[lane], 0.0f);
    h1s[w][lane + 32] = fmaxf(t1 * inv * p.g1[lane + 32] + p.be1[lane + 32], 0.0f);
    __syncthreads();
    // stage 2: 32 dims, LN over 32, relu
    float a2 = p.b2[lane];
    for (int k = 0; k < 64; ++k) a2 += h1s[w][k] * p.W2[lane * 64 + k];
    const float h2 = fmaxf(wave_ln(a2, p.g2[lane], p.be2[lane]), 0.0f);
    // stage 3: scalar head
    const float tot = wsum(h2 * p.W3[lane]) + p.b3[0];
    if (lane == 0)
      out[t * 8 + w] = (t == 0) ? tot : 1.0f / (1.0f + __expf(-tot));
    __syncthreads();
  }
}

// ---------------------------------------------------------------------------
extern "C" void kernel_launch(void* const* d_in, const int* in_sizes, int n_in,
                              void* d_out, int out_size, void* d_ws, size_t ws_size,
                              hipStream_t stream) {
  const float* spatial  = (const float*)d_in[0];
  const float* genomic  = (const float*)d_in[1];
  const float* temporal = (const float*)d_in[2];
  const float* adj      = (const float*)d_in[3];
  // params in dict order, starting at index 4
  const float* spW  = (const float*)d_in[4];
  const float* spb  = (const float*)d_in[5];
  const float* spg  = (const float*)d_in[6];
  const float* spbe = (const float*)d_in[7];
  const float* geW  = (const float*)d_in[8];
  const float* geb  = (const float*)d_in[9];
  const float* geg  = (const float*)d_in[10];
  const float* gebe = (const float*)d_in[11];
  const float* teW  = (const float*)d_in[12];
  const float* teb  = (const float*)d_in[13];
  const float* teg  = (const float*)d_in[14];
  const float* tebe = (const float*)d_in[15];
  const float* gatW  = (const float*)d_in[16];
  const float* gata  = (const float*)d_in[17];
  const float* gatoW = (const float*)d_in[18];
  const float* gatob = (const float*)d_in[19];
  const float* gatg  = (const float*)d_in[20];
  const float* gatbe = (const float*)d_in[21];
  const float* mhaiW = (const float*)d_in[22];
  const float* mhaib = (const float*)d_in[23];
  const float* mhaoW = (const float*)d_in[24];
  const float* mhaob = (const float*)d_in[25];

  TowerP mp { (const float*)d_in[26], (const float*)d_in[27], (const float*)d_in[28],
              (const float*)d_in[29], (const float*)d_in[30], (const float*)d_in[31],
              (const float*)d_in[32], (const float*)d_in[33], (const float*)d_in[34],
              (const float*)d_in[35] };
  TowerP cp { (const float*)d_in[36], (const float*)d_in[37], (const float*)d_in[38],
              (const float*)d_in[39], (const float*)d_in[40], (const float*)d_in[41],
              (const float*)d_in[42], (const float*)d_in[43], (const float*)d_in[44],
              (const float*)d_in[45] };

  float* ws = (float*)d_ws;
  size_t off = 0;
  float* h_ws  = ws + off; off += (size_t)Bv * Nv * 64;    // GAT features
  float* s_ws  = ws + off; off += (size_t)Bv * Nv * Hv;    // src logits
  float* dvals = ws + off; off += (size_t)Bv * Nv * Hv;    // dst logits
  float* hcat  = ws + off; off += (size_t)Bv * Nv * 64;    // aggregated GAT out
  float* q_ws  = ws + off; off += (size_t)Bv * Hv * Nv * HDv;
  float* k_ws  = ws + off; off += (size_t)Bv * Hv * Nv * HDv;
  float* v_ws  = ws + off; off += (size_t)Bv * Hv * Nv * HDv;
  float* o_ws  = ws + off; off += (size_t)Bv * Nv * 32;
  float* rep   = ws + off; off += (size_t)Bv * 32;

  hipMemsetAsync(rep, 0, (size_t)Bv * 32 * sizeof(float), stream);

  k_embed<<<Bv * Nv / 8, 256, 0, stream>>>(
      spatial, genomic, temporal, spW, spb, spg, spbe, geW, geb, geg, gebe,
      teW, teb, teg, tebe, gatW, gata, h_ws, s_ws, dvals);

  k_gat<<<Bv * Hv * (Nv / 64), 128, 0, stream>>>(h_ws, s_ws, dvals, adj, hcat);

  k_gproj<<<Bv * Nv / 8, 256, 0, stream>>>(
      hcat, gatoW, gatob, gatg, gatbe, mhaiW, mhaib, q_ws, k_ws, v_ws);

  k_attn<<<Bv * Hv * (Nv / 64), 128, 0, stream>>>(q_ws, k_ws, v_ws, o_ws);

  k_oproj<<<Bv * Nv / 8, 256, 0, stream>>>(o_ws, mhaoW, mhaob, rep);

  k_tower<<<1, 256, 0, stream>>>(rep, mp, cp, (float*)d_out);

  (void)in_sizes; (void)n_in; (void)out_size; (void)ws_size;
}